// GraphSAGE_29901562315014
// MI455X (gfx1250) — compile-verified
//
#include <hip/hip_runtime.h>
#include <hip/hip_bf16.h>

#define NN 100000
#define IN_F 32
#define HID_F 64
#define OUT_F 16

typedef float v2f __attribute__((ext_vector_type(2)));
typedef float v8f __attribute__((ext_vector_type(8)));

// ---------------- workspace zero ----------------
__global__ __launch_bounds__(256) void zero_ws_kernel(float* __restrict__ ws, long n) {
    long i = (long)blockIdx.x * blockDim.x + threadIdx.x;
    long stride = (long)gridDim.x * blockDim.x;
    for (; i < n; i += stride) ws[i] = 0.0f;
}

// ---------------- edge scatter: layer 1 (32 feats) ----------------
// one wave per edge: lane l adds feat[src][l] into agg[dst][l]; lane0 bumps degree
__global__ __launch_bounds__(256) void scatter32_kernel(const float* __restrict__ feat,
                                                        const int* __restrict__ src,
                                                        const int* __restrict__ dst,
                                                        float* __restrict__ agg,
                                                        float* __restrict__ deg,
                                                        int nE) {
    int tid  = blockIdx.x * blockDim.x + threadIdx.x;
    int e    = tid >> 5;
    int lane = tid & 31;
    if (e >= nE) return;
    int s = src[e];
    int d = dst[e];
    atomicAdd(&agg[(long)d * IN_F + lane], feat[(long)s * IN_F + lane]);
    if (lane == 0) atomicAdd(&deg[d], 1.0f);
}

// ---------------- edge scatter: layer 2 (64 feats) ----------------
__global__ __launch_bounds__(256) void scatter64_kernel(const float* __restrict__ h,
                                                        const int* __restrict__ src,
                                                        const int* __restrict__ dst,
                                                        float* __restrict__ agg,
                                                        int nE) {
    int tid  = blockIdx.x * blockDim.x + threadIdx.x;
    int e    = tid >> 5;
    int lane = tid & 31;
    if (e >= nE) return;
    int s = src[e];
    int d = dst[e];
    atomicAdd(&agg[(long)d * HID_F + lane],      h[(long)s * HID_F + lane]);
    atomicAdd(&agg[(long)d * HID_F + lane + 32], h[(long)s * HID_F + lane + 32]);
}

// ---------------- layer 1 GEMM: h = relu(X@W1s + mean@W1n + b1) ----------------
// one wave = one 16x16 output tile; K=32 in steps of 4 via V_WMMA_F32_16X16X4_F32
__global__ __launch_bounds__(256) void sage_layer1_kernel(const float* __restrict__ feat,
                                                          const float* __restrict__ agg,
                                                          const float* __restrict__ deg,
                                                          const float* __restrict__ Wself,
                                                          const float* __restrict__ Wneigh,
                                                          const float* __restrict__ bias,
                                                          float* __restrict__ h,
                                                          int nRowTiles) {
    int tid  = blockIdx.x * blockDim.x + threadIdx.x;
    int wave = tid >> 5;
    int lane = tid & 31;
    const int colTiles = HID_F / 16;            // 4
    int tileRow = wave / colTiles;
    int ct      = wave % colTiles;
    if (tileRow >= nRowTiles) return;           // whole-wave uniform

    int hi   = lane >> 4;                       // 0 or 1 (half-wave)
    int l15  = lane & 15;
    long row = (long)tileRow * 16 + l15;        // A-operand row (same for both halves)
    int colW = ct * 16 + l15;                   // B/C column

    float dg   = fmaxf(deg[row], 1.0f);
    float invd = 1.0f / dg;

    v8f c;
    float bv = bias[colW];
#pragma unroll
    for (int v = 0; v < 8; ++v) c[v] = bv;

    const float* fr = feat + row * IN_F;
    const float* ar = agg  + row * IN_F;

#pragma unroll
    for (int k0 = 0; k0 < IN_F; k0 += 4) {
        int kA = k0 + 2 * hi;                   // A/B lane-half K base
        v2f a, b;
        // self term
        a.x = fr[kA];
        a.y = fr[kA + 1];
        b.x = Wself[(long)(kA)     * HID_F + colW];
        b.y = Wself[(long)(kA + 1) * HID_F + colW];
        c = __builtin_amdgcn_wmma_f32_16x16x4_f32(false, a, false, b, (short)0, c, false, false);
        // mean-neighbor term (division folded into A load)
        a.x = ar[kA]     * invd;
        a.y = ar[kA + 1] * invd;
        b.x = Wneigh[(long)(kA)     * HID_F + colW];
        b.y = Wneigh[(long)(kA + 1) * HID_F + colW];
        c = __builtin_amdgcn_wmma_f32_16x16x4_f32(false, a, false, b, (short)0, c, false, false);
    }

#pragma unroll
    for (int v = 0; v < 8; ++v) {
        long r = (long)tileRow * 16 + v + 8 * hi;       // C layout: row = v + 8*half
        h[r * HID_F + colW] = fmaxf(c[v], 0.0f);        // ReLU
    }
}

// ---------------- layer 2 GEMM: out = h@W2s + mean2@W2n + b2 ----------------
// one wave = one 16x16 output tile (OUT_F == 16, single column tile); K=64
__global__ __launch_bounds__(256) void sage_layer2_kernel(const float* __restrict__ h,
                                                          const float* __restrict__ agg,
                                                          const float* __restrict__ deg,
                                                          const float* __restrict__ Wself,
                                                          const float* __restrict__ Wneigh,
                                                          const float* __restrict__ bias,
                                                          float* __restrict__ out,
                                                          int nRowTiles) {
    int tid  = blockIdx.x * blockDim.x + threadIdx.x;
    int wave = tid >> 5;
    int lane = tid & 31;
    int tileRow = wave;
    if (tileRow >= nRowTiles) return;

    int hi  = lane >> 4;
    int l15 = lane & 15;
    long row = (long)tileRow * 16 + l15;
    int colW = l15;

    float dg   = fmaxf(deg[row], 1.0f);
    float invd = 1.0f / dg;

    v8f c;
    float bv = bias[colW];
#pragma unroll
    for (int v = 0; v < 8; ++v) c[v] = bv;

    const float* hr = h   + row * HID_F;
    const float* ar = agg + row * HID_F;

#pragma unroll
    for (int k0 = 0; k0 < HID_F; k0 += 4) {
        int kA = k0 + 2 * hi;
        v2f a, b;
        a.x = hr[kA];
        a.y = hr[kA + 1];
        b.x = Wself[(long)(kA)     * OUT_F + colW];
        b.y = Wself[(long)(kA + 1) * OUT_F + colW];
        c = __builtin_amdgcn_wmma_f32_16x16x4_f32(false, a, false, b, (short)0, c, false, false);
        a.x = ar[kA]     * invd;
        a.y = ar[kA + 1] * invd;
        b.x = Wneigh[(long)(kA)     * OUT_F + colW];
        b.y = Wneigh[(long)(kA + 1) * OUT_F + colW];
        c = __builtin_amdgcn_wmma_f32_16x16x4_f32(false, a, false, b, (short)0, c, false, false);
    }

#pragma unroll
    for (int v = 0; v < 8; ++v) {
        long r = (long)tileRow * 16 + v + 8 * hi;
        out[r * OUT_F + colW] = c[v];
    }
}

extern "C" void kernel_launch(void* const* d_in, const int* in_sizes, int n_in,
                              void* d_out, int out_size, void* d_ws, size_t ws_size,
                              hipStream_t stream) {
    const float* feat = (const float*)d_in[0];
    const int*   src  = (const int*)d_in[1];
    const int*   dst  = (const int*)d_in[2];
    const float* W1s  = (const float*)d_in[3];
    const float* W1n  = (const float*)d_in[4];
    const float* b1   = (const float*)d_in[5];
    const float* W2s  = (const float*)d_in[6];
    const float* W2n  = (const float*)d_in[7];
    const float* b2   = (const float*)d_in[8];
    float* out = (float*)d_out;

    const int nNodes = in_sizes[0] / IN_F;   // 100000 (multiple of 16)
    const int nE     = in_sizes[1];          // 1600000
    const int nRowTiles = nNodes / 16;       // 6250

    // workspace layout: deg | agg1[N*32] | agg2[N*64] | h[N*64]
    float* ws   = (float*)d_ws;
    float* deg  = ws;
    float* agg1 = deg  + nNodes;
    float* agg2 = agg1 + (long)nNodes * IN_F;
    float* h    = agg2 + (long)nNodes * HID_F;

    // zero deg + agg1 + agg2 every call (deterministic; h fully overwritten)
    long zn = (long)nNodes * (1 + IN_F + HID_F);
    zero_ws_kernel<<<4096, 256, 0, stream>>>(ws, zn);

    // layer 1: scatter then fused GEMM
    {
        long threads = (long)nE * 32;
        int blocks = (int)((threads + 255) / 256);
        scatter32_kernel<<<blocks, 256, 0, stream>>>(feat, src, dst, agg1, deg, nE);
    }
    {
        int waves = nRowTiles * (HID_F / 16);            // 25000
        int blocks = (waves * 32 + 255) / 256;
        sage_layer1_kernel<<<blocks, 256, 0, stream>>>(feat, agg1, deg, W1s, W1n, b1, h, nRowTiles);
    }

    // layer 2: scatter then fused GEMM
    {
        long threads = (long)nE * 32;
        int blocks = (int)((threads + 255) / 256);
        scatter64_kernel<<<blocks, 256, 0, stream>>>(h, src, dst, agg2, nE);
    }
    {
        int waves = nRowTiles;                           // 6250
        int blocks = (waves * 32 + 255) / 256;
        sage_layer2_kernel<<<blocks, 256, 0, stream>>>(h, agg2, deg, W2s, W2n, b2, out, nRowTiles);
    }
}